// ClassifyModel_80135499809377
// MI455X (gfx1250) — compile-verified
//
#include <hip/hip_runtime.h>

// Problem constants (match the reference)
#define BB    4      // docs
#define SS    8      // sections per doc
#define LSEC  256    // section length incl CLS/SEP
#define LL    254    // stripped section length
#define DD    768    // hidden
#define MAXS  100    // max sentences per doc
#define TT    (SS * LL)   // 2032 tokens per doc

typedef __attribute__((ext_vector_type(2))) float v2f;
typedef __attribute__((ext_vector_type(4))) float f4;
typedef __attribute__((ext_vector_type(8))) float v8f;

// -------------------------------------------------------------------------
// 1) Per-sentence max pooling: pooled[b, m, d] = max over tokens with
//    sentence id == m+1, else 0.  One block per (b, m); ids staged in LDS.
// -------------------------------------------------------------------------
__global__ void __launch_bounds__(256)
maxpool_kernel(const float* __restrict__ atf,   // [B*S, LSEC, D]
               const int*   __restrict__ ids,   // [B, S, LL]
               float*       __restrict__ pooled) // [B, MAXS, D]
{
  const int bm  = blockIdx.x;         // 0 .. B*MAXS-1
  const int b   = bm / MAXS;
  const int m   = bm % MAXS;          // sentence id = m + 1
  const int tid = threadIdx.x;

  __shared__ int sh_ids[LL];

  float a0 = -INFINITY, a1 = -INFINITY, a2 = -INFINITY;

  for (int j = 0; j < SS; ++j) {
    for (int p = tid; p < LL; p += 256) sh_ids[p] = ids[(b * SS + j) * LL + p];
    __syncthreads();
    const float* base = atf + ((size_t)(b * SS + j) * LSEC + 1) * DD; // skip [CLS]
    for (int p = 0; p < LL; ++p) {
      if (sh_ids[p] == m + 1) {
        const float* tok = base + (size_t)p * DD;
        a0 = fmaxf(a0, tok[tid]);
        a1 = fmaxf(a1, tok[tid + 256]);
        a2 = fmaxf(a2, tok[tid + 512]);
      }
    }
    __syncthreads();
  }

  float* out = pooled + (size_t)bm * DD;
  out[tid]       = (a0 == -INFINITY) ? 0.0f : a0;
  out[tid + 256] = (a1 == -INFINITY) ? 0.0f : a1;
  out[tid + 512] = (a2 == -INFINITY) ? 0.0f : a2;
}

// -------------------------------------------------------------------------
// 2) sentence_feature = pooled @ W_g2^T via V_WMMA_F32_16X16X4_F32.
//    One wave32 per 16x16 C tile; K stepped by 4.
//    A (16x4) layout: lanes 0-15 -> A[M=lane][k,k+1]; lanes 16-31 -> [k+2,k+3]
//    B (4x16)  = W_g2^T: B[k][n] = W[n][k], same per-lane k split.
// -------------------------------------------------------------------------
__global__ void __launch_bounds__(32)
gemm_wmma_kernel(const float* __restrict__ A,  // [400, 768]  (pooled)
                 const float* __restrict__ W,  // [768, 768]  (W_g2, row e col d)
                 float*       __restrict__ C)  // [400, 768]
{
  const int TM = (BB * MAXS) / 16;   // 25 row tiles
  const int tile = blockIdx.x;
  const int tm = tile % TM;
  const int tn = tile / TM;
  const int row0 = tm * 16, col0 = tn * 16;

  const int lane = threadIdx.x;
  const int half = lane >> 4;        // 0: k,k+1   1: k+2,k+3
  const int lrow = lane & 15;

  const float* Ap = A + (size_t)(row0 + lrow) * DD;
  const float* Bp = W + (size_t)(col0 + lrow) * DD;

  v8f c = {0.f, 0.f, 0.f, 0.f, 0.f, 0.f, 0.f, 0.f};

  for (int k = 0; k < DD; k += 4) {
    const int kk = k + half * 2;
    v2f a = *(const v2f*)(Ap + kk);
    v2f b = *(const v2f*)(Bp + kk);
    c = __builtin_amdgcn_wmma_f32_16x16x4_f32(
        /*neg_a=*/false, a, /*neg_b=*/false, b,
        /*c_mod=*/(short)0, c, /*reuse_a=*/false, /*reuse_b=*/false);
  }

  // C/D layout: VGPR r -> M = r + half*8, N = lrow
  #pragma unroll
  for (int r = 0; r < 8; ++r)
    C[(size_t)(row0 + r + half * 8) * DD + col0 + lrow] = c[r];
}

// -------------------------------------------------------------------------
// 3) tran_mask [B, T, S]: one-hot of section index, one token row per thread.
// -------------------------------------------------------------------------
__global__ void __launch_bounds__(256)
tran_mask_kernel(float* __restrict__ out)   // [B, T, S]
{
  const int idx = blockIdx.x * 256 + threadIdx.x;  // over B*T
  if (idx >= BB * TT) return;
  const int t = idx % TT;
  const int j = t / LL;
  f4 lo = {0.f, 0.f, 0.f, 0.f}, hi = {0.f, 0.f, 0.f, 0.f};
  if (j < 4) lo[j] = 1.0f; else hi[j - 4] = 1.0f;
  f4* p = (f4*)(out + (size_t)idx * SS);
  p[0] = lo;
  p[1] = hi;
}

// -------------------------------------------------------------------------
// 4) tran_mask_sec [B, MAXS, S]: presence bitmap per section -> counts ->
//    exclusive scan -> interval membership. One block per doc.
// -------------------------------------------------------------------------
__global__ void __launch_bounds__(256)
tran_sec_kernel(const int* __restrict__ ids,   // [B, S, LL]
                float*     __restrict__ out)   // [B, MAXS, S]
{
  const int b = blockIdx.x, tid = threadIdx.x;
  __shared__ unsigned char pres[SS][MAXS + 1];
  __shared__ int counts[SS], starts[SS];

  for (int i = tid; i < SS * (MAXS + 1); i += 256) (&pres[0][0])[i] = 0;
  __syncthreads();

  for (int i = tid; i < SS * LL; i += 256) {
    const int j = i / LL, p = i - j * LL;
    const int id = ids[(b * SS + j) * LL + p];
    if ((unsigned)id <= MAXS) pres[j][id] = 1;
  }
  __syncthreads();

  if (tid < SS) {
    int cnt = 0;
    for (int v = 0; v <= MAXS; ++v) cnt += pres[tid][v];
    counts[tid] = cnt;
  }
  __syncthreads();

  if (tid == 0) {
    int acc = 0;
    for (int j = 0; j < SS; ++j) { starts[j] = acc; acc += counts[j]; }
  }
  __syncthreads();

  for (int i = tid; i < MAXS * SS; i += 256) {
    const int r = i / SS, j = i - r * SS;
    out[(size_t)b * MAXS * SS + i] =
        (r >= starts[j] && r < starts[j] + counts[j]) ? 1.0f : 0.0f;
  }
}

// -------------------------------------------------------------------------
// 5) sentence_mask_full + section_mask_full [B, T, T]: one block per output
//    row; diagonal 254-wide block computed from LDS-staged ids / attn mask,
//    everything else zero.  B128 nontemporal stores (pure streaming output).
// -------------------------------------------------------------------------
__global__ void __launch_bounds__(256)
masks_kernel(const int* __restrict__ ids,       // [B, S, LL]
             const int* __restrict__ am,        // [B*S, LSEC]
             float*     __restrict__ sent_out,  // [B, T, T]
             float*     __restrict__ sec_out)   // [B, T, T]
{
  const int gb = blockIdx.x;          // b * T + i
  const int b  = gb / TT;
  const int i  = gb % TT;
  const int j  = i / LL;
  const int pi = i - j * LL;
  const int tid = threadIdx.x;

  __shared__ int   sh_ids[LL];
  __shared__ float sh_am[LL];
  for (int p = tid; p < LL; p += 256) {
    sh_ids[p] = ids[(b * SS + j) * LL + p];
    sh_am[p]  = (float)am[(b * SS + j) * LSEC + p + 1];
  }
  __syncthreads();

  const int   my_id = sh_ids[pi];
  const float my_am = sh_am[pi];
  const int lo = j * LL, hi = lo + LL;

  f4* srow = (f4*)(sent_out + (size_t)gb * TT);
  f4* crow = (f4*)(sec_out  + (size_t)gb * TT);

  for (int q = tid; q < TT / 4; q += 256) {
    f4 vs = {0.f, 0.f, 0.f, 0.f};
    f4 vc = {0.f, 0.f, 0.f, 0.f};
    const int k0 = q * 4;
    if (k0 + 3 >= lo && k0 < hi) {
      #pragma unroll
      for (int c = 0; c < 4; ++c) {
        const int k = k0 + c;
        if (k >= lo && k < hi) {
          const int kp = k - lo;
          vs[c] = (sh_ids[kp] == my_id) ? 1.0f : 0.0f;
          vc[c] = my_am * sh_am[kp];
        }
      }
    }
    __builtin_nontemporal_store(vs, srow + q);
    __builtin_nontemporal_store(vc, crow + q);
  }
}

// -------------------------------------------------------------------------
// 6) logits1 = classifier(max over sections of pooled_feature). 8 outputs.
// -------------------------------------------------------------------------
__global__ void __launch_bounds__(256)
logits_kernel(const float* __restrict__ pf,   // [B*S, D]
              const float* __restrict__ Wc,   // [2, D]
              const float* __restrict__ bc,   // [2]
              float*       __restrict__ out)  // [B, 2]
{
  const int b = blockIdx.x >> 1, cls = blockIdx.x & 1, tid = threadIdx.x;
  __shared__ float red[256];
  float acc = 0.0f;
  for (int d = tid; d < DD; d += 256) {
    float m = pf[(size_t)(b * SS) * DD + d];
    for (int j = 1; j < SS; ++j) m = fmaxf(m, pf[(size_t)(b * SS + j) * DD + d]);
    acc += m * Wc[cls * DD + d];
  }
  red[tid] = acc;
  __syncthreads();
  for (int s = 128; s > 0; s >>= 1) {
    if (tid < s) red[tid] += red[tid + s];
    __syncthreads();
  }
  if (tid == 0) out[b * 2 + cls] = red[0] + bc[cls];
}

// -------------------------------------------------------------------------
extern "C" void kernel_launch(void* const* d_in, const int* in_sizes, int n_in,
                              void* d_out, int out_size, void* d_ws, size_t ws_size,
                              hipStream_t stream)
{
  const float* atf = (const float*)d_in[0];   // all_token_feature [32,256,768] f32
  const float* pf  = (const float*)d_in[1];   // pooled_feature    [32,768]     f32
  const int*   am  = (const int*)  d_in[2];   // attention_mask    [32,256]     i32
  const int*   ids = (const int*)  d_in[3];   // sentence_mask     [4,8,254]    i32
  const float* Wg2 = (const float*)d_in[4];   // W_g2 [768,768]
  const float* Wc  = (const float*)d_in[5];   // W_cls [2,768]
  const float* bc  = (const float*)d_in[6];   // b_cls [2]
  float* out = (float*)d_out;

  float* pooled = (float*)d_ws;               // [400,768] f32 scratch, rewritten each call

  // Output tuple offsets (flat f32 elements, reference return order)
  constexpr size_t OFF_SF  = 0;                             // sentence_feature  [4,100,768]
  constexpr size_t OFF_TM  = OFF_SF  + (size_t)BB * MAXS * DD;   // 307200
  constexpr size_t OFF_TMS = OFF_TM  + (size_t)BB * TT * SS;     // 372224
  constexpr size_t OFF_SMF = OFF_TMS + (size_t)BB * MAXS * SS;   // 375424
  constexpr size_t OFF_SEC = OFF_SMF + (size_t)BB * TT * TT;     // 16891520
  constexpr size_t OFF_LG  = OFF_SEC + (size_t)BB * TT * TT;     // 33407616

  maxpool_kernel<<<BB * MAXS, 256, 0, stream>>>(atf, ids, pooled);
  gemm_wmma_kernel<<<((BB * MAXS) / 16) * (DD / 16), 32, 0, stream>>>(pooled, Wg2, out + OFF_SF);
  tran_mask_kernel<<<(BB * TT + 255) / 256, 256, 0, stream>>>(out + OFF_TM);
  tran_sec_kernel<<<BB, 256, 0, stream>>>(ids, out + OFF_TMS);
  masks_kernel<<<BB * TT, 256, 0, stream>>>(ids, am, out + OFF_SMF, out + OFF_SEC);
  logits_kernel<<<BB * 2, 256, 0, stream>>>(pf, Wc, bc, out + OFF_LG);
}